// ForwardModelSimple_91242285236610
// MI455X (gfx1250) — compile-verified
//
#include <hip/hip_runtime.h>

typedef __attribute__((ext_vector_type(2))) float v2f;
typedef __attribute__((ext_vector_type(8))) float v8f;

__device__ __forceinline__ float pick8(const v8f& d, int w) {
    switch (w) {
        case 0: return d[0]; case 1: return d[1];
        case 2: return d[2]; case 3: return d[3];
        case 4: return d[4]; case 5: return d[5];
        case 6: return d[6]; default: return d[7];
    }
}

__global__ __launch_bounds__(256) void ForwardModelSimple_91242285236610_kernel(
    const float* __restrict__ vol,     // (256,256,256) z,y,x
    const float* __restrict__ quat,    // (B,4)
    float* __restrict__ out)           // (B,256,256)
{
    const int b    = blockIdx.y;
    const int tile = blockIdx.x;           // 0..255 -> 16x16 tiles over (i,j)
    const int i0   = (tile >> 4) << 4;
    const int j0   = (tile & 15) << 4;

    // ---- quaternion -> rotation matrix (uniform per block) ----
    float qw = quat[b * 4 + 0], qx = quat[b * 4 + 1];
    float qy = quat[b * 4 + 2], qz = quat[b * 4 + 3];
    float nrm = sqrtf(qw * qw + qx * qx + qy * qy + qz * qz);
    qw /= nrm; qx /= nrm; qy /= nrm; qz /= nrm;
    const float R00 = 1.f - 2.f * (qy * qy + qz * qz);
    const float R01 = 2.f * (qx * qy - qz * qw);
    const float R02 = 2.f * (qx * qz + qy * qw);
    const float R10 = 2.f * (qx * qy + qz * qw);
    const float R11 = 1.f - 2.f * (qx * qx + qz * qz);
    const float R12 = 2.f * (qy * qz - qx * qw);
    const float R20 = 2.f * (qx * qz - qy * qw);
    const float R21 = 2.f * (qy * qz + qx * qw);
    const float R22 = 1.f - 2.f * (qx * qx + qy * qy);

    // ---- thread <-> pixel mapping matched to WMMA D layout ----
    const int t  = threadIdx.x;
    const int l  = t & 31;            // lane in wave32
    const int w  = t >> 5;            // wave id 0..7
    const int tj = l & 15;
    const int ti = w + ((l >> 4) << 3);   // VGPR w holds M=w (lanes 0-15), M=w+8 (lanes 16-31)

    // ---- WMMA 16x16x4 computes p0 = R*(i-128, j-128, -128) for the whole tile ----
    // A(m,0)=i0+m-128, A(m,1)=1, A(m,2..3)=0 ; B_c(0,n)=Rc0, B_c(1,n)=Rc1*(j0+n-128)-128*Rc2
    const bool  lo16 = (l < 16);
    const float gi = (float)(i0 + (l & 15)) - 128.f;
    const float gj = (float)(j0 + (l & 15)) - 128.f;

    v2f a;  a.x  = lo16 ? gi  : 0.f;  a.y  = lo16 ? 1.f : 0.f;
    v2f bx; bx.x = lo16 ? R00 : 0.f;  bx.y = lo16 ? (R01 * gj - 128.f * R02) : 0.f;
    v2f by; by.x = lo16 ? R10 : 0.f;  by.y = lo16 ? (R11 * gj - 128.f * R12) : 0.f;
    v2f bz; bz.x = lo16 ? R20 : 0.f;  bz.y = lo16 ? (R21 * gj - 128.f * R22) : 0.f;
    v8f cz = {};

    v8f dx = __builtin_amdgcn_wmma_f32_16x16x4_f32(false, a, false, bx, (short)0, cz, false, false);
    v8f dy = __builtin_amdgcn_wmma_f32_16x16x4_f32(false, a, false, by, (short)0, cz, false, false);
    v8f dz = __builtin_amdgcn_wmma_f32_16x16x4_f32(false, a, false, bz, (short)0, cz, false, false);

    const float p0x = pick8(dx, w);
    const float p0y = pick8(dy, w);
    const float p0z = pick8(dz, w);

    // ---- walk the line: p(k) = p0 + k * R[:,2], trilinear sample, accumulate ----
    const float S = 255.f / 256.f;      // exact in binary
    float acc = 0.f;

    for (int k = 0; k < 256; ++k) {
        const float kf = (float)k;
        float px = fmaf(kf, R02, p0x);
        float py = fmaf(kf, R12, p0y);
        float pz = fmaf(kf, R22, p0z);
        // clip(rg,-1,1) <=> clamp(p,-128,128); pix = (p+128)*255/256
        px = fminf(fmaxf(px, -128.f), 128.f);
        py = fminf(fmaxf(py, -128.f), 128.f);
        pz = fminf(fmaxf(pz, -128.f), 128.f);
        const float gxp = (px + 128.f) * S;   // indexes x (last axis)
        const float gyp = (py + 128.f) * S;   // indexes y
        const float gzp = (pz + 128.f) * S;   // indexes z (first axis)

        const float fx = floorf(gxp), fy = floorf(gyp), fz = floorf(gzp);
        const float tx = gxp - fx,   ty = gyp - fy,   tz = gzp - fz;
        const int   x0 = (int)fx,    y0 = (int)fy,    z0 = (int)fz;   // in [0,255]

        const int   xb = min(x0, 254);
        const float sx = tx + (float)(x0 - xb);     // ==1 at x-edge (tx==0 there)
        const int   dy_ = (y0 < 255) ? 256 : 0;
        const int   dz_ = (z0 < 255) ? 65536 : 0;

        const float* p00 = vol + ((z0 << 16) + (y0 << 8) + xb);
        const float* p01 = p00 + dy_;
        const float* p10 = p00 + dz_;
        const float* p11 = p10 + dy_;

        const float a00 = p00[0], b00 = p00[1];
        const float a01 = p01[0], b01 = p01[1];
        const float a10 = p10[0], b10 = p10[1];
        const float a11 = p11[0], b11 = p11[1];

        const float c00 = fmaf(sx, b00 - a00, a00);
        const float c01 = fmaf(sx, b01 - a01, a01);
        const float c10 = fmaf(sx, b10 - a10, a10);
        const float c11 = fmaf(sx, b11 - a11, a11);
        const float c0  = fmaf(ty, c01 - c00, c00);
        const float c1  = fmaf(ty, c11 - c10, c10);
        acc += fmaf(tz, c1 - c0, c0);
    }

    out[(b << 16) + ((i0 + ti) << 8) + (j0 + tj)] = acc;
}

extern "C" void kernel_launch(void* const* d_in, const int* in_sizes, int n_in,
                              void* d_out, int out_size, void* d_ws, size_t ws_size,
                              hipStream_t stream) {
    const float* vol  = (const float*)d_in[0];   // (256^3) float32
    const float* quat = (const float*)d_in[1];   // (B,4)  float32
    float* out = (float*)d_out;                  // (B,256,256) float32
    const int B = in_sizes[1] / 4;

    dim3 grid(256, B);   // 256 tiles of 16x16 over (i,j), one tile-row per batch
    dim3 block(256);     // 8 wave32 = one 16x16 pixel tile
    ForwardModelSimple_91242285236610_kernel<<<grid, block, 0, stream>>>(vol, quat, out);
}